// UnsupervisedMorphemeSegmenter_40089224741403
// MI455X (gfx1250) — compile-verified
//
#include <hip/hip_runtime.h>

// UnsupervisedMorphemeSegmenter, fused single-kernel version for MI455X
// (gfx1250, wave32). Forward value reduces to:
//   scores[b,c] = we[b,c,:].W + bias  (masked)
//   top-(nm-1) separators -> char->morpheme one-hot bpm
//   me[b] = bpm^T (12x64) x we[b] (64x768)     [exact f32 via WMMA 16x16x4]
// One block per batch: we[b] (192KB) is staged once into CDNA5's 320KB LDS
// via GLOBAL_LOAD_ASYNC_TO_LDS_B128, so HBM sees each input byte exactly
// once (~0.48GB total => ~21us at 23.3TB/s vs ~38us for the 2-pass version).

#define NEGV (-1000000000.0f)

static constexpr int Cc  = 64;          // chars per word (T)
static constexpr int Mm  = 12;          // max morphemes
static constexpr int Hh  = 768;         // hidden
static constexpr int PAD = 784;         // LDS row stride (dwords): 784%64=16,
                                        // 2*784%64=32 -> WMMA B reads conflict-free
static constexpr int WE_OFF = 0;                 // we tile: 64 x PAD floats
static constexpr int S_OFF  = Cc * PAD;          // S matrix: 64 x 16 floats
static constexpr int SC_OFF = S_OFF + Cc * 16;   // scores:   64 floats
static constexpr int SMEM_DW = SC_OFF + Cc;      // total dwords (51264 -> ~200KB)

typedef float v2f __attribute__((ext_vector_type(2)));
typedef float v8f __attribute__((ext_vector_type(8)));
typedef int   v4i __attribute__((ext_vector_type(4)));
typedef __attribute__((address_space(1))) v4i gv4i;   // global int4
typedef __attribute__((address_space(3))) v4i lv4i;   // LDS int4

#if defined(__has_builtin)
#if __has_builtin(__builtin_amdgcn_global_load_async_to_lds_b128) && \
    __has_builtin(__builtin_amdgcn_s_wait_asynccnt)
#define USE_ASYNC_LDS 1
#endif
#endif

__global__ void seg_fused(const float* __restrict__ we,
                          const int* __restrict__ wl_p,
                          const int* __restrict__ nm_p,
                          const float* __restrict__ W,
                          const float* __restrict__ bias,
                          float* __restrict__ me,
                          float* __restrict__ bpm_out) {
  extern __shared__ float smem[];
  const int b    = blockIdx.x;
  const int tid  = threadIdx.x;          // 256 threads = 8 waves
  const int wave = tid >> 5;
  const int lane = tid & 31;

  // ---------------- Phase 0: stage we[b] (64x768 f32) into LDS -------------
  const float* __restrict__ gsrc = we + (size_t)b * Cc * Hh;
  for (int j = tid; j < (Cc * Hh) / 4; j += 256) {   // 48 x 16B per thread
    const int row = j / (Hh / 4);
    const int c4  = j - row * (Hh / 4);
    const int ldsoff = WE_OFF + row * PAD + c4 * 4;
#ifdef USE_ASYNC_LDS
    __builtin_amdgcn_global_load_async_to_lds_b128(
        (gv4i*)(gsrc + (size_t)j * 4),     // global int4* (addrspace 1)
        (lv4i*)(&smem[ldsoff]),            // LDS int4*    (addrspace 3)
        /*offset=*/0, /*cpol=*/0);
#else
    const float4 v = *(const float4*)(gsrc + (size_t)j * 4);
    *(float4*)(&smem[ldsoff]) = v;
#endif
  }
#ifdef USE_ASYNC_LDS
  __builtin_amdgcn_s_wait_asynccnt(0);
#endif
  __syncthreads();

  const int wlb = wl_p[b];
  const int nmb = nm_p[b];

  // ---------------- Phase 1: scores from LDS (8 chars per wave) ------------
  {
    const float bsc = bias[0];
#pragma unroll
    for (int cc = 0; cc < 8; ++cc) {
      const int c = wave * 8 + cc;
      float acc = 0.0f;
#pragma unroll
      for (int t = 0; t < Hh / 32; ++t)
        acc += smem[WE_OFF + c * PAD + lane + 32 * t] * W[lane + 32 * t];
#pragma unroll
      for (int off = 16; off >= 1; off >>= 1)
        acc += __shfl_xor(acc, off, 32);
      if (lane == 0) {
        float v = acc + bsc;
        if (c >= wlb - 1) v = NEGV;     // score_mask (last split col masked)
        smem[SC_OFF + c] = v;
      }
    }
  }
  __syncthreads();

  // ---------------- Phase 2 (wave 0): top-k -> c2m -> S + bpm --------------
  if (wave == 0) {
    float v0 = smem[SC_OFF + lane];       // char c = lane
    float v1 = smem[SC_OFF + 32 + lane];  // char c = lane + 32
    const int nsep = nmb - 1;
    unsigned f0 = 0u, f1 = 0u;            // separator flags per char

    for (int it = 0; it < nsep; ++it) {   // uniform trip count within wave
      float bv; int bi;                   // max value, ties -> lower index
      if (v0 >= v1) { bv = v0; bi = lane; } else { bv = v1; bi = lane + 32; }
#pragma unroll
      for (int off = 16; off >= 1; off >>= 1) {
        float ov = __shfl_xor(bv, off, 32);
        int   oi = __shfl_xor(bi, off, 32);
        if (ov > bv || (ov == bv && oi < bi)) { bv = ov; bi = oi; }
      }
      if (bi == lane)      v0 = -3.0e38f;  // remove winner
      if (bi == lane + 32) v1 = -3.0e38f;
      const int p = (bi + 1) & 63;         // separator position (mirrors roll)
      if (p == lane)      f0 = 1u;
      if (p == lane + 32) f1 = 1u;
    }

#if __has_builtin(__builtin_amdgcn_ballot_w32)
    const unsigned m0 = __builtin_amdgcn_ballot_w32(f0 != 0u);
    const unsigned m1 = __builtin_amdgcn_ballot_w32(f1 != 0u);
#else
    const unsigned m0 = (unsigned)__ballot(f0 != 0u);
    const unsigned m1 = (unsigned)__ballot(f1 != 0u);
#endif
    const unsigned incl = 0xFFFFFFFFu >> (31 - lane);  // bits 0..lane
    int c2m0 = __popc(m0 & incl);                       // inclusive prefix
    int c2m1 = __popc(m0) + __popc(m1 & incl);
    if (lane >= wlb)      c2m0 = 255;                   // padded chars
    if (lane + 32 >= wlb) c2m1 = 255;

    float* __restrict__ o0 = bpm_out + ((size_t)b * Cc + lane) * Mm;
    float* __restrict__ o1 = o0 + 32 * Mm;
#pragma unroll
    for (int m = 0; m < 16; ++m) {        // S padded M: rows 12..15 zero
      const float s0 = (c2m0 == m) ? 1.0f : 0.0f;
      const float s1 = (c2m1 == m) ? 1.0f : 0.0f;
      smem[S_OFF + lane * 16 + m]        = s0;
      smem[S_OFF + (lane + 32) * 16 + m] = s1;
      if (m < Mm) { o0[m] = s0; o1[m] = s1; }           // bpm output (0/1)
    }
  }
  __syncthreads();

  // ---------------- Phase 3: me[b] = S(16x64) x we[b](64x768), f32 WMMA ----
  const int g  = lane >> 4;               // lane half selects K pair
  const int ln = lane & 15;               // M (for A) / N (for B,D)

  for (int tile = wave; tile < Hh / 16; tile += 8) {    // 6 tiles per wave
    const int n0 = tile * 16;
    v8f acc = {0.f, 0.f, 0.f, 0.f, 0.f, 0.f, 0.f, 0.f};
#pragma unroll
    for (int k = 0; k < Cc; k += 4) {
      // A 16x4 f32: VGPR0 = K=(k+2g), VGPR1 = K=(k+1+2g), lane%16 = M
      v2f a, bb;
      a.x = smem[S_OFF + (k + 2 * g) * 16 + ln];
      a.y = smem[S_OFF + (k + 1 + 2 * g) * 16 + ln];
      // B 4x16 f32 from padded LDS tile (lane halves hit disjoint banks)
      bb.x = smem[WE_OFF + (k + 2 * g) * PAD + n0 + ln];
      bb.y = smem[WE_OFF + (k + 1 + 2 * g) * PAD + n0 + ln];
      acc = __builtin_amdgcn_wmma_f32_16x16x4_f32(
          /*neg_a=*/false, a, /*neg_b=*/false, bb,
          /*c_mod=*/(short)0, acc, /*reuse_a=*/false, /*reuse_b=*/false);
    }
    // D 16x16 f32: lanes 0-15 -> M=r (r=0..7), lanes 16-31 -> M=8+r.
    // Rows m>=nm are exact zeros (S rows zero) and overwrite 0xAA poison.
    float* __restrict__ outb = me + ((size_t)b * Mm) * Hh + n0 + ln;
    if (g == 0) {
#pragma unroll
      for (int r = 0; r < 8; ++r) outb[(size_t)r * Hh] = acc[r];
    } else {
#pragma unroll
      for (int r = 0; r < 4; ++r) outb[(size_t)(8 + r) * Hh] = acc[r];
    }
  }
}

extern "C" void kernel_launch(void* const* d_in, const int* in_sizes, int n_in,
                              void* d_out, int out_size, void* d_ws, size_t ws_size,
                              hipStream_t stream) {
  const float* we   = (const float*)d_in[0];   // (B,64,768) f32
  const int*   wl   = (const int*)d_in[1];     // (B,) i32
  const int*   nm   = (const int*)d_in[2];     // (B,) i32
  const float* W    = (const float*)d_in[3];   // (768,) f32
  const float* bias = (const float*)d_in[4];   // scalar f32

  const int B = in_sizes[1];                   // 2048

  float* me  = (float*)d_out;                              // (B*12, 768)
  float* bpm = (float*)d_out + (size_t)B * Mm * Hh;        // (B, 64, 12)

  const size_t smem_bytes = (size_t)SMEM_DW * sizeof(float);   // ~200KB LDS
  seg_fused<<<B, 256, smem_bytes, stream>>>(we, wl, nm, W, bias, me, bpm);
}